// MCPBRNN_Generic_PETconstraint_constantoutput_variableLoss_MCA2_10196252360843
// MI455X (gfx1250) — compile-verified
//
#include <hip/hip_runtime.h>
#include <stdint.h>

// MCPBRNN scalar recurrence (hidden_size == 1), B = 8192, T = 1.
// Phase A (1 block): std(y_obs[SPIN:TRAIN], ddof=1) -> ws[0]
// Phase B (1 block): TDM x -> LDS, parallel ol[] precompute, serial scan by
//                    thread 0 with algebraic fast path, parallel outputs.

#define ML_C   2.9086f
#define SL_C   1.898f
#define SPIN_C 365
#define TRAIN_C 4000
#define NB 8192

typedef unsigned int v4u  __attribute__((ext_vector_type(4)));
typedef int          v8i_ __attribute__((ext_vector_type(8)));
typedef int          v4i_ __attribute__((ext_vector_type(4)));

__device__ __forceinline__ float fast_exp(float x) {
  return __builtin_amdgcn_exp2f(x * 1.44269504088896340736f);   // v_exp_f32
}
__device__ __forceinline__ float fast_rcp(float x) {
  return __builtin_amdgcn_rcpf(x);                              // v_rcp_f32
}
__device__ __forceinline__ float fast_sigmoid(float z) {
  return fast_rcp(1.0f + fast_exp(-z));
}

// ---------------- Phase A: unbiased std of y_obs[SPIN:TRAIN] ----------------
__global__ __launch_bounds__(256) void stats_kernel(const float* __restrict__ y,
                                                    float* __restrict__ ws) {
  __shared__ float ssum[256];
  __shared__ float ssq[256];
  const int tid = threadIdx.x;
  float s = 0.0f, q = 0.0f;
  for (int i = SPIN_C + tid; i < TRAIN_C; i += 256) {
    const float v = y[i];
    s += v;
    q = __builtin_fmaf(v, v, q);
  }
  ssum[tid] = s;
  ssq[tid]  = q;
  __syncthreads();
  for (int off = 128; off > 0; off >>= 1) {
    if (tid < off) {
      ssum[tid] += ssum[tid + off];
      ssq[tid]  += ssq[tid + off];
    }
    __syncthreads();
  }
  if (tid == 0) {
    const float n    = (float)(TRAIN_C - SPIN_C);
    const float mean = ssum[0] / n;
    const float var  = (ssq[0] - mean * ssum[0]) / (n - 1.0f);
    ws[0] = sqrtf(fmaxf(var, 0.0f));
  }
}

// ---------------- Phase B: TDM stage + serial scan + parallel outputs -------
__global__ __launch_bounds__(1024) void mcpbrnn_kernel(
    const float* __restrict__ x,       // (NB, 1, 2) interleaved u1,u2
    const float* __restrict__ w_yom, const float* __restrict__ w_gw,
    const float* __restrict__ w_lm,  const float* __restrict__ w_fm,
    const float* __restrict__ b0p,   const float* __restrict__ wb2p,
    const int*   __restrict__ tlagp, const float* __restrict__ ws,
    float* __restrict__ out) {
  __shared__ float sx[2 * NB];   // 64 KB: staged x (TDM)
  __shared__ float sol[NB];      // 32 KB: precomputed loss gate ol[b]
  __shared__ float scin[NB];     // 32 KB: entry carry c_b per step

  const int tid = threadIdx.x;

  // Uniform gate constants (redundant per-thread scalar work, negligible).
  const float e_om = fast_exp(w_yom[0]);
  const float e_gw = fast_exp(w_gw[0]);
  const float e_lm = fast_exp(w_lm[0]);
  const float e_fm = fast_exp(w_fm[0]);
  const float rden = fast_rcp(e_om + e_gw + e_lm + e_fm);
  const float oo   = e_om * rden;      // constant output gate
  const float oogw = e_gw * rden;      // constant groundwater gate
  const float ol1  = e_lm * rden;      // loss-gate scale
  const float b0   = b0p[0];
  const float wb2  = wb2p[0];
  const float K    = 1.0f - oo - oogw; // f = relu(K - olc_raw); K > ol always
  const int tlag   = tlagp[0];

  // --- TDM: DMA whole x (16384 f32 = 64 KB) into LDS; wave 0 issues it. ---
  if (tid < 32) {
    const unsigned lds_off = (unsigned)(unsigned long long)(void*)&sx[0];
    const unsigned long long ga = (unsigned long long)(const void*)x;
    // D# group 0: count=1 | lds_addr | global_addr(57b) | type=2
    v4u g0 = { 1u,
               lds_off,
               (unsigned)(ga & 0xFFFFFFFFull),
               (unsigned)((ga >> 32) & 0x01FFFFFFull) | (2u << 30) };
    // D# group 1: data_size=4B; tensor_dim0=16384, tensor_dim1=1;
    //             tile_dim0=16384, tile_dim1=1; dim0_stride=16384.
    enum { N0 = 2 * NB };
    v8i_ g1 = { (int)(2u << 16),                          // data_size=2 (4B)
                (int)(((unsigned)N0 & 0xFFFFu) << 16),    // dim0[15:0] @63:48
                (int)(((unsigned)N0 >> 16) | (1u << 16)), // dim0[31:16] | dim1 lo
                (int)((unsigned)N0 << 16),                // dim1 hi(0) | tile_dim0
                1,                                        // tile_dim1=1, tile_dim2=0
                (int)N0,                                  // dim0_stride lo32
                0, 0 };
    v4i_ gz = { 0, 0, 0, 0 };
#if defined(__clang_major__) && (__clang_major__ >= 23)
    v8i_ gz8 = { 0, 0, 0, 0, 0, 0, 0, 0 };
    __builtin_amdgcn_tensor_load_to_lds(g0, g1, gz, gz, gz8, 0);
#else
    __builtin_amdgcn_tensor_load_to_lds(g0, g1, gz, gz, 0);
#endif
    __builtin_amdgcn_s_wait_tensorcnt(0);
  }
  __syncthreads();

  // --- Parallel precompute: ol[b] (sigmoid off the serial critical path). ---
  for (int i = tid; i < NB; i += 1024) {
    const float u2 = sx[2 * i + 1];
    sol[i] = ol1 * fast_sigmoid(b0 + (u2 - ML_C) * (1.0f / SL_C) * wb2);
  }
  __syncthreads();

  // --- Serial nonlinear recurrence: single thread.
  // Fast path (c>0, u2 <= ol*c): elu(z)=z  =>  olc_raw=u2/c, f=K-u2/c>0
  //   =>  c' = K*c - u2 + u1   (no rcp, no exp on the dependent chain).
  // Slow path (c>0, u2 > ol*c): full  olc_raw = ol + 1 - exp(ol - u2/c).
  // c<=0 path: olc_raw = ol.
  if (tid == 0) {
    float c = 0.0f;
    for (int b = 0; b < NB; ++b) {
      scin[b] = c;                       // entry carry (outputs use this)
      const float u1 = sx[2 * b + 0];
      const float u2 = sx[2 * b + 1];
      const float ol = sol[b];
      if (b >= tlag) {
        if (c > 0.0f) {
          const float olc = ol * c;
          if (u2 <= olc) {
            c = (K * c - u2) + u1;                       // fast path
          } else {
            const float z       = ol - u2 * fast_rcp(c);
            const float olc_raw = (ol + 1.0f) - fast_exp(z);
            const float f       = fmaxf(K - olc_raw, 0.0f);
            c = __builtin_fmaf(f, c, u1);
          }
        } else {
          const float f = fmaxf(K - ol, 0.0f);
          c = __builtin_fmaf(f, c, u1);
        }
      }
    }
  }
  __syncthreads();

  // --- Parallel per-step outputs (coalesced, bandwidth-bound). ---
  const float obsstd = ws[0];
  for (int i = tid; i < NB; i += 1024) {
    const float c  = scin[i];
    const float u2 = sx[2 * i + 1];
    const float ol = sol[i];
    const float t  = u2 * fast_rcp(c);
    const float z  = ol - t;
    const float el = (z > 0.0f) ? z : (fast_exp(z) - 1.0f);   // elu
    const float olc_raw = (c > 0.0f) ? (ol - el) : ol;
    const float olc = fmaxf(olc_raw, 0.0f);
    const float f   = fmaxf(K - olc_raw, 0.0f);
    const bool  dm  = (i >= tlag);

    const float hv = dm ? oo * c : 0.0f;
    out[0  * NB + i] = hv;                      // h_n
    out[1  * NB + i] = dm ? c         : 0.0f;   // c_n
    out[2  * NB + i] = dm ? ol * c    : 0.0f;   // l_n
    out[3  * NB + i] = dm ? olc * c   : 0.0f;   // lc_n
    out[4  * NB + i] = dm ? oogw * c  : 0.0f;   // gw_n
    out[5  * NB + i] = 0.0f;                    // bp_n
    out[6  * NB + i] = 0.0f;                    // Gate_ib
    out[7  * NB + i] = dm ? oo        : 0.0f;   // g_oo
    out[8  * NB + i] = dm ? oogw      : 0.0f;   // g_oogw
    out[9  * NB + i] = dm ? ol        : 0.0f;   // g_ol
    out[10 * NB + i] = dm ? olc       : 0.0f;   // g_olc
    out[11 * NB + i] = dm ? f         : 0.0f;   // g_f
    const float ob = dm ? obsstd : 0.0f;
    out[12 * NB + 2 * i + 0] = hv;              // h_nout[:,0]
    out[12 * NB + 2 * i + 1] = ob;              // h_nout[:,1]
    out[14 * NB + i] = ob;                      // obs_std
  }
}

extern "C" void kernel_launch(void* const* d_in, const int* in_sizes, int n_in,
                              void* d_out, int out_size, void* d_ws, size_t ws_size,
                              hipStream_t stream) {
  const float* x     = (const float*)d_in[0];
  const float* y_obs = (const float*)d_in[1];
  const float* wyom  = (const float*)d_in[2];
  const float* wgw   = (const float*)d_in[3];
  const float* wlm   = (const float*)d_in[4];
  const float* wfm   = (const float*)d_in[5];
  const float* b0    = (const float*)d_in[6];
  const float* wb2   = (const float*)d_in[7];
  // d_in[8] = epoch (unused)
  const int* tlag    = (const int*)d_in[9];
  float* out = (float*)d_out;
  float* ws  = (float*)d_ws;

  stats_kernel<<<1, 256, 0, stream>>>(y_obs, ws);
  mcpbrnn_kernel<<<1, 1024, 0, stream>>>(x, wyom, wgw, wlm, wfm, b0, wb2,
                                         tlag, ws, out);
}